// MirasMemoryLayer_6528350290243
// MI455X (gfx1250) — compile-verified
//
#include <hip/hip_runtime.h>
#include <math.h>

typedef __attribute__((ext_vector_type(2))) float v2f;
typedef __attribute__((ext_vector_type(8))) float v8f;

#define DIM   512
#define HEADS 8
#define HD    64
#define T_LEN 1024
#define LRNK  32

// ---------------------------------------------------------------------------
// f32 WMMA GEMM:  C[m,n] = ACT( sum_k (A[m,k] * Amul[m,k] if GATED) * B[n,k] )
// A: (M,K) row-major, B: (N,K) row-major  (computes A @ B^T).
// One wave per 16x32 output macro-tile (two V_WMMA_F32_16X16X4_F32 accums,
// shared A fragment). K-loop unrolled x4 -> 8 WMMAs per iteration.
// ACT: 0 = none, 1 = SiLU, 2 = sigmoid.  M%16==0, N%32==0, K%16==0.
// ---------------------------------------------------------------------------
template <int ACT, bool GATED>
__global__ void wmma_gemm_f32_kernel(const float* __restrict__ A,
                                     const float* __restrict__ B,
                                     const float* __restrict__ Amul,
                                     float* __restrict__ C,
                                     int M, int N, int Kd)
{
    const int wave    = (int)(threadIdx.x >> 5);
    const int lane    = (int)(threadIdx.x & 31);
    const int tiles_n = N >> 5;                       // 32-wide column tiles
    const int total   = (M >> 4) * tiles_n;
    const int tile    = blockIdx.x * 8 + wave;
    if (tile >= total) return;

    const int tm = (tile / tiles_n) << 4;
    const int tn = (tile % tiles_n) << 5;

    const int half = lane >> 4;       // selects K sub-pair within the 4-step
    const int l15  = lane & 15;       // M (A) / N (B) index within 16
    const int ka   = half * 2;

    const float* arow  = A + (size_t)(tm + l15) * Kd + ka;
    const float* grow  = GATED ? (Amul + (size_t)(tm + l15) * Kd + ka) : nullptr;
    const float* brow0 = B + (size_t)(tn + l15) * Kd + ka;
    const float* brow1 = B + (size_t)(tn + 16 + l15) * Kd + ka;

    v8f acc0 = {};
    v8f acc1 = {};

    for (int k = 0; k < Kd; k += 16) {
        v2f a[4], b0[4], b1[4];
#pragma unroll
        for (int u = 0; u < 4; ++u) {
            const int kk = k + 4 * u;
            v2f av = *(const v2f*)(arow + kk);
            if (GATED) {
                v2f gv = *(const v2f*)(grow + kk);
                av[0] *= gv[0];
                av[1] *= gv[1];
            }
            a[u]  = av;
            b0[u] = *(const v2f*)(brow0 + kk);
            b1[u] = *(const v2f*)(brow1 + kk);
        }
#pragma unroll
        for (int u = 0; u < 4; ++u) {
            acc0 = __builtin_amdgcn_wmma_f32_16x16x4_f32(
                false, a[u], false, b0[u], (short)0, acc0, false, false);
            acc1 = __builtin_amdgcn_wmma_f32_16x16x4_f32(
                false, a[u], false, b1[u], (short)0, acc1, false, false);
        }
    }

    // C/D layout: VGPR v -> M = v (lanes 0-15) or v+8 (lanes 16-31); N = l15.
    const int col0 = tn + l15;
    const int col1 = tn + 16 + l15;
#pragma unroll
    for (int v = 0; v < 8; ++v) {
        const int row = tm + v + half * 8;
        float y0 = acc0[v];
        float y1 = acc1[v];
        if (ACT == 1) {                       // SiLU
            y0 = y0 / (1.0f + __expf(-y0));
            y1 = y1 / (1.0f + __expf(-y1));
        } else if (ACT == 2) {                // sigmoid
            y0 = 1.0f / (1.0f + __expf(-y0));
            y1 = 1.0f / (1.0f + __expf(-y1));
        }
        C[(size_t)row * N + col0] = y0;
        C[(size_t)row * N + col1] = y1;
    }
}

// ---------------------------------------------------------------------------
// Fused causal depthwise conv (K=4) for k & v streams + per-head RMSNorm on k.
// One block per timestep, one thread per channel.
// ---------------------------------------------------------------------------
__global__ void conv_norm_kernel(const float* __restrict__ kpre,
                                 const float* __restrict__ vpre,
                                 const float* __restrict__ ckw,
                                 const float* __restrict__ ckb,
                                 const float* __restrict__ cvw,
                                 const float* __restrict__ cvb,
                                 const float* __restrict__ knw,
                                 float* __restrict__ k_seq,
                                 float* __restrict__ v_seq)
{
    const int t = blockIdx.x;
    const int c = threadIdx.x;

    float kk = ckb[c];
    float vv = cvb[c];
#pragma unroll
    for (int j = 0; j < 4; ++j) {
        const int tt = t - 3 + j;
        if (tt >= 0) {
            kk += kpre[(size_t)tt * DIM + c] * ckw[c * 4 + j];
            vv += vpre[(size_t)tt * DIM + c] * cvw[c * 4 + j];
        }
    }
    v_seq[(size_t)t * DIM + c] = vv;

    __shared__ float sq[DIM];
    __shared__ float rinv[HEADS];
    sq[c] = kk * kk;
    __syncthreads();
    if ((c & (HD - 1)) == 0) {
        float sum = 0.0f;
        for (int i = 0; i < HD; ++i) sum += sq[c + i];
        rinv[c >> 6] = rsqrtf(sum * (1.0f / (float)HD) + 1e-6f);
    }
    __syncthreads();
    k_seq[(size_t)t * DIM + c] = kk * rinv[c >> 6] * knw[c & (HD - 1)];
}

// ---------------------------------------------------------------------------
// Sequential state scan. One wave32 per state row; each lane keeps 16 state
// columns in registers (no LDS, no cross-wave traffic). Two lane reductions
// per step via shfl_xor. Writes per-step outputs and the final state.
// ---------------------------------------------------------------------------
__global__ void scan_kernel(const float* __restrict__ k_seq,
                            const float* __restrict__ v_seq,
                            const float* __restrict__ eta,
                            const float* __restrict__ alpha,
                            float* __restrict__ outputs,
                            float* __restrict__ final_state)
{
    const int lane = (int)(threadIdx.x & 31);
    const int row  = (int)((blockIdx.x * blockDim.x + threadIdx.x) >> 5);
    if (row >= DIM) return;
    const int c0 = lane << 4;   // 16 columns per lane

    float s[16];
#pragma unroll
    for (int i = 0; i < 16; ++i) s[i] = 0.0f;

    for (int t = 0; t < T_LEN; ++t) {
        // This lane's 64B slice of k_t (L2-resident; all waves share k_seq)
        float4 k4[4];
        const float4* kp = (const float4*)(k_seq + (size_t)t * DIM + c0);
#pragma unroll
        for (int q = 0; q < 4; ++q) k4[q] = kp[q];
        const float* kv = (const float*)k4;

        // err_row = dot(S[row,:], k_t) - v_t[row]
        float err = 0.0f;
#pragma unroll
        for (int i = 0; i < 16; ++i) err = fmaf(s[i], kv[i], err);
#pragma unroll
        for (int off = 16; off >= 1; off >>= 1) err += __shfl_xor(err, off, 32);
        err -= v_seq[(size_t)t * DIM + row];

        const float a    = alpha[(size_t)t * DIM + row];
        const float coef = eta[(size_t)t * DIM + row] * err;

        // S = clip(a*S - coef * k^T, +-10);  out_row = dot(S_new[row,:], k_t)
        float o = 0.0f;
#pragma unroll
        for (int i = 0; i < 16; ++i) {
            float ns = fmaf(a, s[i], -coef * kv[i]);
            ns = fminf(10.0f, fmaxf(-10.0f, ns));
            s[i] = ns;
            o = fmaf(ns, kv[i], o);
        }
#pragma unroll
        for (int off = 16; off >= 1; off >>= 1) o += __shfl_xor(o, off, 32);
        if (lane == 0) outputs[(size_t)t * DIM + row] = o;
    }

    float4* fs = (float4*)(final_state + (size_t)row * DIM + c0);
    const float4* sv = (const float4*)s;
#pragma unroll
    for (int q = 0; q < 4; ++q) fs[q] = sv[q];
}

// ---------------------------------------------------------------------------
extern "C" void kernel_launch(void* const* d_in, const int* in_sizes, int n_in,
                              void* d_out, int out_size, void* d_ws, size_t ws_size,
                              hipStream_t stream)
{
    const float* x   = (const float*)d_in[0];
    const float* Wk  = (const float*)d_in[1];
    const float* Wv  = (const float*)d_in[2];
    const float* Wo  = (const float*)d_in[3];
    const float* Wg  = (const float*)d_in[4];
    const float* ckw = (const float*)d_in[5];
    const float* ckb = (const float*)d_in[6];
    const float* cvw = (const float*)d_in[7];
    const float* cvb = (const float*)d_in[8];
    const float* knw = (const float*)d_in[9];
    const float* ed  = (const float*)d_in[10];
    const float* eu  = (const float*)d_in[11];
    const float* ad  = (const float*)d_in[12];
    const float* au  = (const float*)d_in[13];

    float* ws = (float*)d_ws;
    const size_t TD = (size_t)T_LEN * DIM;
    const size_t TL = (size_t)T_LEN * LRNK;
    float* k_pre = ws;
    float* v_pre = k_pre + TD;
    float* gate  = v_pre + TD;
    float* emid  = gate  + TD;
    float* amid  = emid  + TL;
    float* etab  = amid  + TL;
    float* alph  = etab  + TD;
    float* k_seq = alph  + TD;
    float* v_seq = k_seq + TD;
    float* outs  = v_seq + TD;

    float* proj   = (float*)d_out;          // (T, DIM)
    float* fstate = proj + TD;              // (DIM, DIM)

    const dim3 blk(256);
    auto nblocks = [](int M, int N) { return ((M >> 4) * (N >> 5) + 7) / 8; };

    // Projections (full f32 WMMA path, branch-free hot loops).
    wmma_gemm_f32_kernel<0, false><<<nblocks(T_LEN, DIM), blk, 0, stream>>>(x, Wk, nullptr, k_pre, T_LEN, DIM, DIM);
    wmma_gemm_f32_kernel<0, false><<<nblocks(T_LEN, DIM), blk, 0, stream>>>(x, Wv, nullptr, v_pre, T_LEN, DIM, DIM);
    wmma_gemm_f32_kernel<1, false><<<nblocks(T_LEN, DIM), blk, 0, stream>>>(x, Wg, nullptr, gate,  T_LEN, DIM, DIM);
    // Low-rank eta / alpha with fused sigmoid on the up-projection.
    wmma_gemm_f32_kernel<0, false><<<nblocks(T_LEN, LRNK), blk, 0, stream>>>(x, ed, nullptr, emid, T_LEN, LRNK, DIM);
    wmma_gemm_f32_kernel<2, false><<<nblocks(T_LEN, DIM),  blk, 0, stream>>>(emid, eu, nullptr, etab, T_LEN, DIM, LRNK);
    wmma_gemm_f32_kernel<0, false><<<nblocks(T_LEN, LRNK), blk, 0, stream>>>(x, ad, nullptr, amid, T_LEN, LRNK, DIM);
    wmma_gemm_f32_kernel<2, false><<<nblocks(T_LEN, DIM),  blk, 0, stream>>>(amid, au, nullptr, alph, T_LEN, DIM, LRNK);

    // Causal depthwise conv + per-head RMSNorm.
    conv_norm_kernel<<<T_LEN, DIM, 0, stream>>>(k_pre, v_pre, ckw, ckb, cvw, cvb, knw, k_seq, v_seq);

    // Row-parallel sequential scan: 512 waves, state in registers.
    scan_kernel<<<(DIM * 32) / 256, 256, 0, stream>>>(k_seq, v_seq, etab, alph, outs, fstate);

    // Output projection with fused elementwise gate on the A operand.
    wmma_gemm_f32_kernel<0, true><<<nblocks(T_LEN, DIM), blk, 0, stream>>>(outs, Wo, gate, proj, T_LEN, DIM, DIM);
}